// EmbeddingSimilarityModel_24266565222679
// MI455X (gfx1250) — compile-verified
//
#include <hip/hip_runtime.h>
#include <hip/hip_bf16.h>

// Problem constants (fixed by the reference).
#define NN 16384      // number of vectors / group-id space
#define DD 256        // embedding dim
#define BB 64         // batch size of the sequential scan
#define WORDS_PER_ROW (NN / 32)   // 512 u32 words of match bits per row

#define KC     32     // K-chunk staged in LDS per pipeline stage
#define NCHUNK (DD / KC)

typedef float v2f __attribute__((ext_vector_type(2)));
typedef float v8f __attribute__((ext_vector_type(8)));

// ---------------------------------------------------------------------------
// Tensor Data Mover: 2D tile load (tileRows x tileK f32) from a row-major
// matrix with row stride DD into LDS (rows stored contiguously, [row][k]).
// D# layout per CDNA5 ISA 8.3-8.6. Arity differs between toolchains.
// ---------------------------------------------------------------------------
#if defined(__has_builtin)
#if __has_builtin(__builtin_amdgcn_tensor_load_to_lds)
#define HAVE_TDM 1
#endif
#endif

#ifdef HAVE_TDM
#warning CDNA5_PATH: TDM tensor_load_to_lds staging ENABLED
typedef unsigned int u32x4 __attribute__((ext_vector_type(4)));
typedef int          i32x4 __attribute__((ext_vector_type(4)));
typedef int          i32x8 __attribute__((ext_vector_type(8)));

__device__ __forceinline__ void tdm_load_tile(unsigned ldsAddr, const float* gptr,
                                              int tileK, int tileRows)
{
    unsigned long long ga = (unsigned long long)(size_t)gptr;
    // Group 0: count=1 (valid), lds_addr, global_addr[56:0], type=2 ("image").
    u32x4 g0 = { 1u,
                 ldsAddr,
                 (unsigned)(ga & 0xFFFFFFFFull),
                 (unsigned)((ga >> 32) & 0x01FFFFFFull) | (2u << 30) };
    // Group 1: data_size=4B(code 2); tensor_dim0=DD; tensor_dim1=NN;
    // tile_dim0=tileK; tile_dim1=tileRows; tensor_dim0_stride=DD.
    i32x8 g1 = { (int)0x00020000,
                 (int)((DD & 0xFFFF) << 16),   // tensor_dim0 lo16
                 (int)((NN & 0xFFFF) << 16),   // tensor_dim0 hi=0 | tensor_dim1 lo16
                 (int)(tileK << 16),           // tensor_dim1 hi=0 | tile_dim0
                 tileRows,                     // tile_dim1 (tile_dim2 = 0)
                 DD,                           // tensor_dim0_stride low 32 (= 256)
                 0, 0 };
    i32x4 z4 = {0, 0, 0, 0};
#if __clang_major__ >= 23
    i32x8 z8 = {0, 0, 0, 0, 0, 0, 0, 0};
    __builtin_amdgcn_tensor_load_to_lds(g0, g1, z4, z4, z8, 0);
#else
    __builtin_amdgcn_tensor_load_to_lds(g0, g1, z4, z4, 0);
#endif
}
#else
#warning CDNA5_PATH: TDM builtin NOT available - global->LDS copy fallback
#endif // HAVE_TDM

// ---------------------------------------------------------------------------
// WMMA compute for one K-chunk out of LDS. Per wave: 16x64 output strip,
// 4 f32 accumulators, A reused across the 4 column sub-tiles.
// Ab: [128][KC] (block rows), Bb: [64][KC] (block cols).
// ---------------------------------------------------------------------------
__device__ __forceinline__ void compute_chunk(const float* Ab, const float* Bb,
                                              int aOff, int bOff, v8f acc[4])
{
    #pragma unroll
    for (int kk = 0; kk < KC; kk += 4) {
        v2f a = *(const v2f*)(Ab + aOff + kk);
        #pragma unroll
        for (int t = 0; t < 4; ++t) {
            v2f b = *(const v2f*)(Bb + bOff + t * 16 * KC + kk);
            acc[t] = __builtin_amdgcn_wmma_f32_16x16x4_f32(
                false, a, false, b, (short)0, acc[t], false, false);
        }
    }
}

// ---------------------------------------------------------------------------
// Kernel 1: full Gram matrix V @ V^T in fp32 on the WMMA path, thresholded and
// packed to a bit matrix: bits[i][j] = (dot(Vi,Vj) >= thr) && (j >= (i/64)*64+1)
//
// Block = 256 threads = 8 waves; block tile = 128 rows x 64 cols. A/B panels
// staged in LDS per K-chunk; TDM double-buffers chunk c+1 while chunk c is
// consumed by V_WMMA_F32_16X16X4_F32 from ds_load operands.
// ---------------------------------------------------------------------------
__global__ __launch_bounds__(256)
void score_bits_kernel(const float* __restrict__ V,
                       unsigned* __restrict__ bits,
                       const float* __restrict__ thr_p)
{
    __shared__ __align__(128) float Abuf[2][128 * KC];  // 2 x 16 KB
    __shared__ __align__(128) float Bbuf[2][64 * KC];   // 2 x  8 KB

    const int lane = threadIdx.x & 31;
    const int wave = threadIdx.x >> 5;            // 0..7
    const int half = lane >> 4;                   // 0: K pair (k0,k0+1), 1: (k0+2,k0+3)
    const int l16  = lane & 15;

    const int blockRow = blockIdx.y * 128;
    const int rowTile  = blockRow + wave * 16;    // 16 output rows for this wave
    const int colBase  = blockIdx.x * 64;         // 64 output cols for this block

    // LDS operand offsets (element index; byte offset is 8B aligned: KC even).
    const int aOff = (wave * 16 + l16) * KC + 2 * half;
    const int bOff = l16 * KC + 2 * half;

    v8f acc[4] = {};

#ifdef HAVE_TDM
    // Prologue: DMA chunk 0 into buffer 0 (one wave drives the TDM).
    if (wave == 0) {
        tdm_load_tile((unsigned)(size_t)&Abuf[0][0], V + (size_t)blockRow * DD, KC, 128);
        tdm_load_tile((unsigned)(size_t)&Bbuf[0][0], V + (size_t)colBase  * DD, KC, 64);
    }
    for (int c = 0; c < NCHUNK; ++c) {
        const int cur = c & 1, nxt = cur ^ 1;
        if (wave == 0) {
            if (c + 1 < NCHUNK) {
                const int k0 = (c + 1) * KC;
                tdm_load_tile((unsigned)(size_t)&Abuf[nxt][0],
                              V + (size_t)blockRow * DD + k0, KC, 128);
                tdm_load_tile((unsigned)(size_t)&Bbuf[nxt][0],
                              V + (size_t)colBase * DD + k0, KC, 64);
                __builtin_amdgcn_s_wait_tensorcnt(2);  // chunk c's 2 DMAs done
            } else {
                __builtin_amdgcn_s_wait_tensorcnt(0);  // last chunk done
            }
        }
        __syncthreads();                               // LDS buf[cur] ready
        compute_chunk(&Abuf[cur][0], &Bbuf[cur][0], aOff, bOff, acc);
        __syncthreads();                               // done reading buf[cur]
    }
#else
    // Fallback: cooperative synchronous staging (single buffer).
    for (int c = 0; c < NCHUNK; ++c) {
        __syncthreads();   // previous chunk fully consumed
        const int k0 = c * KC;
        const float* gA = V + (size_t)blockRow * DD + k0;
        for (int i = threadIdx.x; i < 128 * (KC / 4); i += 256) {
            int r = i / (KC / 4), q = i % (KC / 4);
            *(float4*)&Abuf[0][r * KC + q * 4] = *(const float4*)(gA + (size_t)r * DD + q * 4);
        }
        const float* gB = V + (size_t)colBase * DD + k0;
        for (int i = threadIdx.x; i < 64 * (KC / 4); i += 256) {
            int r = i / (KC / 4), q = i % (KC / 4);
            *(float4*)&Bbuf[0][r * KC + q * 4] = *(const float4*)(gB + (size_t)r * DD + q * 4);
        }
        __syncthreads();
        compute_chunk(&Abuf[0][0], &Bbuf[0][0], aOff, bOff, acc);
    }
#endif

    // Threshold + column mask + bit-pack. C/D layout: lane l, vgpr r holds
    // element (row = rowTile + r + 8*(l>=16), col = colBase + t*16 + (l&15)).
    // Ballot: bits[15:0] -> row rowTile+r, bits[31:16] -> row rowTile+r+8.
    // Each u16 slot of the bit matrix is written by exactly one lane grid-wide.
    const float thr = *thr_p;
    #pragma unroll
    for (int t = 0; t < 4; ++t) {
        #pragma unroll
        for (int r = 0; r < 8; ++r) {
            const int m = rowTile + r + 8 * half;        // this lane's global row
            const int n = colBase + t * 16 + l16;        // this lane's global col
            const int rowBatchStart = (m >> 6) << 6;     // batch start for row m
            const bool pred = (acc[t][r] >= thr) && (n >= rowBatchStart + 1);
            const unsigned mask = (unsigned)__ballot((int)pred);
            if (lane == 0) {
                unsigned short* p =
                    (unsigned short*)(bits + (size_t)(rowTile + r) * WORDS_PER_ROW);
                p[(colBase + t * 16) >> 4] = (unsigned short)(mask & 0xFFFFu);
            } else if (lane == 16) {
                unsigned short* p =
                    (unsigned short*)(bits + (size_t)(rowTile + r + 8) * WORDS_PER_ROW);
                p[(colBase + t * 16) >> 4] = (unsigned short)(mask >> 16);
            }
        }
    }
}

// ---------------------------------------------------------------------------
// Kernel 2: the sequential group-merge scan. One workgroup (1024 threads)
// replays the reference exactly:
//   for b in 0..255: g_snap = g; g_i = g_snap[start:start+64]
//     for k in 0..63:
//       mark[gid] = OR over { bits[row][j] && g_i[k] != g_snap[j] } at gid=g_snap[j]
//       g[p] = g_i[k] wherever mark[g[p]]
// Rows are processed in strictly increasing global order, so the next row's
// bit words are prefetched while the current row is handled.
// ---------------------------------------------------------------------------
__global__ __launch_bounds__(1024)
void merge_kernel(const unsigned* __restrict__ bits,
                  const int* __restrict__ g_in,
                  int* __restrict__ g,        // working + final output (d_out)
                  int* __restrict__ g_snap)   // scratch snapshot (d_ws)
{
    __shared__ unsigned mark[WORDS_PER_ROW];  // 2 KB group-mark bitmap
    __shared__ int gi_s[BB];

    const int tid = threadIdx.x;
    const int nthr = blockDim.x;

    for (int p = tid; p < NN; p += nthr) g[p] = g_in[p];
    __syncthreads();

    for (int b = 0; b < NN / BB; ++b) {
        const int start = b * BB;

        // Snapshot g at batch start.
        for (int p = tid; p < NN; p += nthr) g_snap[p] = g[p];
        __syncthreads();
        if (tid < BB) gi_s[tid] = g_snap[start + tid];
        __syncthreads();

        for (int k = 0; k < BB; ++k) {
            // Clear the group-mark bitmap.
            if (tid < WORDS_PER_ROW) mark[tid] = 0u;
            __syncthreads();

            const unsigned* rowbits = bits + (size_t)(start + k) * WORDS_PER_ROW;
            const int gik = gi_s[k];

            // Prefetch next row's bits (2 KB = 16 x 128B lines).
            if (tid < 16)
                __builtin_prefetch((const char*)rowbits + WORDS_PER_ROW * 4 +
                                   (size_t)tid * 128, 0, 3);

            // Sparse scatter: mark[g_snap[j]] for each matched j.
            for (int w = tid; w < WORDS_PER_ROW; w += nthr) {
                unsigned mw = rowbits[w];
                while (mw) {
                    const int bit = __ffs(mw) - 1;
                    mw &= mw - 1u;
                    const int j = (w << 5) + bit;
                    const int gj = g_snap[j];
                    if (gj != gik)
                        atomicOr(&mark[gj >> 5], 1u << (gj & 31));
                }
            }
            __syncthreads();

            // Relabel every position whose current group is marked.
            for (int p = tid; p < NN; p += nthr) {
                const int gp = g[p];
                if ((mark[gp >> 5] >> (gp & 31)) & 1u) g[p] = gik;
            }
            __syncthreads();
        }
    }
}

// ---------------------------------------------------------------------------
// Launch: inputs are (V [N*D f32], group_ids [N i32], cos_threshold [1 f32]).
// d_ws layout: [0, 32MB) match-bit matrix, [32MB, 32MB+64KB) g_snap.
// d_out: final group ids (N int32).
// ---------------------------------------------------------------------------
extern "C" void kernel_launch(void* const* d_in, const int* in_sizes, int n_in,
                              void* d_out, int out_size, void* d_ws, size_t ws_size,
                              hipStream_t stream) {
    const float* V     = (const float*)d_in[0];
    const int*   gids  = (const int*)d_in[1];
    const float* thr_p = (const float*)d_in[2];

    unsigned* bits   = (unsigned*)d_ws;
    int*      g_snap = (int*)((char*)d_ws + (size_t)NN * (NN / 8)); // 32 MB offset
    int*      out    = (int*)d_out;

    dim3 grid(NN / 64, NN / 128);   // col strips x row tiles-of-128
    score_bits_kernel<<<grid, 256, 0, stream>>>(V, bits, thr_p);
    merge_kernel<<<1, 1024, 0, stream>>>(bits, gids, out, g_snap);
}